// GraphSAGE_13683765805695
// MI455X (gfx1250) — compile-verified
//
#include <hip/hip_runtime.h>
#include <math.h>

#define NNODES 100000
#define NEDGES 1600000

typedef __attribute__((ext_vector_type(2))) float v2f;
typedef __attribute__((ext_vector_type(8))) float v8f;

// ---------------------------------------------------------------- degree ----
__global__ void degree_kernel(const int* __restrict__ dst, float* __restrict__ deg, int E) {
    int i = blockIdx.x * blockDim.x + threadIdx.x;
    if (i < E) atomicAdd(&deg[dst[i]], 1.0f);
}

__global__ void invdeg_kernel(float* __restrict__ deg, int n) {
    int i = blockIdx.x * blockDim.x + threadIdx.x;
    if (i < n) deg[i] = 1.0f / fmaxf(deg[i], 1.0f);
}

// ------------------------------------------------------- edge scatter-add ----
// One thread per (edge, 4-float chunk). float4 gather from source row,
// 4x global_atomic_add_f32 into agg[dst] (agg fits in L2: <= 51 MB).
__global__ void scatter_kernel(const float* __restrict__ h, const int* __restrict__ src,
                               const int* __restrict__ dst, float* __restrict__ agg,
                               int E, int din, int chunkShift) {
    int tid = blockIdx.x * blockDim.x + threadIdx.x;
    int chunks = din >> 2;
    if (tid >= E * chunks) return;
    int e = tid >> chunkShift;
    int c = (tid & (chunks - 1)) << 2;
    int s = src[e], d = dst[e];
    const float4 v = *(const float4*)(h + (size_t)s * din + c);
    float* a = agg + (size_t)d * din + c;
    atomicAdd(a + 0, v.x);
    atomicAdd(a + 1, v.y);
    atomicAdd(a + 2, v.z);
    atomicAdd(a + 3, v.w);
}

// --------------------------------------------- WMMA fp32 node update --------
// out[n,:] = (agg[n,:] * invdeg[n]) @ Wl + bl + h[n,:] @ Wr   (+ optional relu)
// One wave -> one 16x16 output tile via V_WMMA_F32_16X16X4_F32.
// 8 waves/block share one node-tile stripe in x, one 16-col weight panel (LDS).
__global__ __launch_bounds__(256) void sage_update_wmma(
    const float* __restrict__ h, const float* __restrict__ agg,
    const float* __restrict__ invdeg,
    const float* __restrict__ Wl, const float* __restrict__ bl,
    const float* __restrict__ Wr, float* __restrict__ out,
    int nTiles, int din, int dout, int applyRelu) {

    __shared__ float sWl[128 * 16];   // weight panel W[:, j0:j0+16], zero-padded
    __shared__ float sWr[128 * 16];

    const int j0 = blockIdx.y * 16;
    for (int idx = threadIdx.x; idx < din * 16; idx += blockDim.x) {
        int k = idx >> 4, j = idx & 15;
        int jj = j0 + j;
        float wl = 0.0f, wr = 0.0f;
        if (jj < dout) { wl = Wl[k * dout + jj]; wr = Wr[k * dout + jj]; }
        sWl[idx] = wl;
        sWr[idx] = wr;
    }
    __syncthreads();

    const int wave = threadIdx.x >> 5;
    const int t = blockIdx.x * 8 + wave;       // 16-node tile index (wave-uniform)
    if (t >= nTiles) return;

    const int lane = threadIdx.x & 31;
    const int half = lane >> 4;                // K-half select for A/B fragments
    const int mn   = lane & 15;                // A-row (m) == B/C-column (n)
    const int row  = t * 16 + mn;              // node index for this lane's A row
    const float inv = invdeg[row];

    // seed accumulator with the bias (broadcast down the M dimension)
    v8f acc;
    const float bb = (j0 + mn < dout) ? bl[j0 + mn] : 0.0f;
#pragma unroll
    for (int r = 0; r < 8; ++r) acc[r] = bb;

    const float* hrow = h   + (size_t)row * din;
    const float* arow = agg + (size_t)row * din;

    for (int k = 0; k < din; k += 4) {
        const int kA = k + 2 * half;           // this lane's K pair
        v2f aR = *(const v2f*)(hrow + kA);     // root-term A fragment
        v2f aA = *(const v2f*)(arow + kA);     // mean-agg A fragment
        aA.x *= inv; aA.y *= inv;
        v2f bR = { sWr[kA * 16 + mn], sWr[(kA + 1) * 16 + mn] };
        v2f bL = { sWl[kA * 16 + mn], sWl[(kA + 1) * 16 + mn] };
        acc = __builtin_amdgcn_wmma_f32_16x16x4_f32(false, aR, false, bR,
                                                    (short)0, acc, false, false);
        acc = __builtin_amdgcn_wmma_f32_16x16x4_f32(false, aA, false, bL,
                                                    (short)0, acc, false, false);
    }

    const int j = j0 + mn;
    if (j < dout) {
#pragma unroll
        for (int r = 0; r < 8; ++r) {
            float v = acc[r];
            if (applyRelu) v = fmaxf(v, 0.0f);
            out[(size_t)(t * 16 + r + 8 * half) * dout + j] = v;
        }
    }
}

// ------------------------------------------------------- log_softmax(40) ----
__global__ void log_softmax40(float* __restrict__ out, int n) {
    int i = blockIdx.x * blockDim.x + threadIdx.x;
    if (i >= n) return;
    float* row = out + (size_t)i * 40;
    float v[40];
    float mx = -INFINITY;
#pragma unroll
    for (int j = 0; j < 40; ++j) { v[j] = row[j]; mx = fmaxf(mx, v[j]); }
    float s = 0.0f;
#pragma unroll
    for (int j = 0; j < 40; ++j) s += __expf(v[j] - mx);
    const float lse = mx + __logf(s);
#pragma unroll
    for (int j = 0; j < 40; ++j) row[j] = v[j] - lse;
}

// ----------------------------------------------------------------- launch ----
extern "C" void kernel_launch(void* const* d_in, const int* in_sizes, int n_in,
                              void* d_out, int out_size, void* d_ws, size_t ws_size,
                              hipStream_t stream) {
    const int N = NNODES, E = NEDGES;
    const float* x   = (const float*)d_in[0];
    const int*   ei  = (const int*)d_in[1];
    const int*   src = ei;            // edge_index[0]
    const int*   dst = ei + E;        // edge_index[1]
    const float* Wl0 = (const float*)d_in[2];
    const float* bl0 = (const float*)d_in[3];
    const float* Wr0 = (const float*)d_in[4];
    const float* Wl1 = (const float*)d_in[5];
    const float* bl1 = (const float*)d_in[6];
    const float* Wr1 = (const float*)d_in[7];
    const float* Wl2 = (const float*)d_in[8];
    const float* bl2 = (const float*)d_in[9];
    const float* Wr2 = (const float*)d_in[10];
    float* outp = (float*)d_out;

    // workspace layout (~128.4 MB)
    float* invdeg = (float*)d_ws;                 // N
    float* agg    = invdeg + N;                   // N*128 (max din)
    float* h1     = agg + (size_t)N * 128;        // N*64
    float* h2     = h1  + (size_t)N * 64;         // N*128

    const int nTiles = (N + 15) / 16;             // 6250 (exact)
    const int gx = (nTiles + 7) / 8;

    // degrees once: the graph is identical across all three layers
    hipMemsetAsync(invdeg, 0, (size_t)N * sizeof(float), stream);
    degree_kernel<<<(E + 255) / 256, 256, 0, stream>>>(dst, invdeg, E);
    invdeg_kernel<<<(N + 255) / 256, 256, 0, stream>>>(invdeg, N);

    // ---- layer 0: 64 -> 64, relu
    hipMemsetAsync(agg, 0, (size_t)N * 64 * sizeof(float), stream);
    scatter_kernel<<<(E * 16 + 255) / 256, 256, 0, stream>>>(x, src, dst, agg, E, 64, 4);
    sage_update_wmma<<<dim3(gx, 4), 256, 0, stream>>>(x, agg, invdeg, Wl0, bl0, Wr0,
                                                      h1, nTiles, 64, 64, 1);

    // ---- layer 1: 64 -> 128, relu
    hipMemsetAsync(agg, 0, (size_t)N * 64 * sizeof(float), stream);
    scatter_kernel<<<(E * 16 + 255) / 256, 256, 0, stream>>>(h1, src, dst, agg, E, 64, 4);
    sage_update_wmma<<<dim3(gx, 8), 256, 0, stream>>>(h1, agg, invdeg, Wl1, bl1, Wr1,
                                                      h2, nTiles, 64, 128, 1);

    // ---- layer 2: 128 -> 40, no relu, straight into d_out
    hipMemsetAsync(agg, 0, (size_t)N * 128 * sizeof(float), stream);
    scatter_kernel<<<(E * 32 + 255) / 256, 256, 0, stream>>>(h2, src, dst, agg, E, 128, 5);
    sage_update_wmma<<<dim3(gx, 3), 256, 0, stream>>>(h2, agg, invdeg, Wl2, bl2, Wr2,
                                                      outp, nTiles, 128, 40, 0);

    // ---- log_softmax over the 40 classes, in place
    log_softmax40<<<(N + 255) / 256, 256, 0, stream>>>(outp, N);
}